// KernalHead_89653147337072
// MI455X (gfx1250) — compile-verified
//
#include <hip/hip_runtime.h>

typedef __attribute__((ext_vector_type(2))) float v2f;
typedef __attribute__((ext_vector_type(8))) float v8f;

#define NEG_SLOPE 0.02f

// ---------------------------------------------------------------------------
// Prep kernel 1: pack conv weights into WMMA A-fragment order.
// Consumer reads ap[((ci*3+ky)*4 + t)*32 + lane] as v2f:
//   m = t*16 + (lane&15); kx0 = (lane<16)?0:2
//   .x = w[m][ci][ky][kx0]; .y = (lane<16)? w[m][ci][ky][1] : 0   (K=3 pad = 0)
// ---------------------------------------------------------------------------
__global__ __launch_bounds__(256) void pack_conv_w(const float* __restrict__ w,
                                                   float* __restrict__ ap) {
  int idx = blockIdx.x * 256 + threadIdx.x;          // 64*3*4*32 = 24576
  if (idx >= 64 * 3 * 4 * 32) return;
  int lane = idx & 31;
  int t    = (idx >> 5) & 3;
  int ky   = (idx >> 7) % 3;
  int ci   = (idx >> 7) / 3;
  int m    = t * 16 + (lane & 15);
  int kx0  = (lane < 16) ? 0 : 2;
  const float* wb = w + ((m * 64 + ci) * 3 + ky) * 3;
  ap[idx * 2 + 0] = wb[kx0];
  ap[idx * 2 + 1] = (lane < 16) ? wb[1] : 0.0f;
}

// ---------------------------------------------------------------------------
// Prep kernel 2: pack pred_w (24 x 768), M padded to 32 with zeros.
// Consumer reads ap[((step*2 + t)*32 + lane)] as v2f, step in [0,192):
//   j0 = step*4; m = t*16+(lane&15); kk = (lane<16)?0:2
//   .x = pw[m][j0+kk]; .y = pw[m][j0+kk+1]; zero if m >= 24
// ---------------------------------------------------------------------------
__global__ __launch_bounds__(256) void pack_pred_w(const float* __restrict__ pw,
                                                   float* __restrict__ ap) {
  int idx = blockIdx.x * 256 + threadIdx.x;          // 192*2*32 = 12288
  if (idx >= 192 * 2 * 32) return;
  int lane = idx & 31;
  int t    = (idx >> 5) & 1;
  int step = idx >> 6;
  int m    = t * 16 + (lane & 15);
  int j    = step * 4 + ((lane < 16) ? 0 : 2);
  float x = 0.0f, y = 0.0f;
  if (m < 24) { x = pw[m * 768 + j]; y = pw[m * 768 + j + 1]; }
  ap[idx * 2 + 0] = x;
  ap[idx * 2 + 1] = y;
}

// ---------------------------------------------------------------------------
// Kernel 1: 3x3 conv + bias + LeakyReLU, implicit GEMM via V_WMMA_F32_16X16X4_F32.
// One block per (b, y) row: 8 waves x 16 pixels, each wave computes all 64 co.
// h written NHWC: h[b][y][x][c], c contiguous.
// ---------------------------------------------------------------------------
__global__ __launch_bounds__(256) void conv_lrelu(const float* __restrict__ x,
                                                  const float* __restrict__ conv_b,
                                                  const float* __restrict__ apack,
                                                  float* __restrict__ h) {
  __shared__ float Xs[192 * 132];                    // 64ci * 3rows * 132cols ~99KB
  const int b = blockIdx.x >> 7;
  const int y = blockIdx.x & 127;
  const int tid = threadIdx.x;

  // Stage input slab with conv zero-padding materialized (cols 0 and 129 are
  // gx=-1 / gx=128; rows outside image are zero; cols 130..131 zeroed so the
  // padded-K B reads are finite).
  for (int i = tid; i < 192 * 132; i += 256) {
    int c  = i % 132;
    int p  = i / 132;
    int ci = p / 3;
    int r  = p - ci * 3;
    int gy = y + r - 1;
    int gx = c - 1;
    float v = 0.0f;
    if ((unsigned)gy < 128u && (unsigned)gx < 128u && c < 130)
      v = x[((b * 64 + ci) * 128 + gy) * 128 + gx];
    Xs[i] = v;
  }
  __syncthreads();

  const int lane = tid & 31;
  const int wave = tid >> 5;
  const int n    = lane & 15;          // pixel within tile
  const int half = lane >> 4;          // 0: K={0,1}  1: K={2,3(pad)}
  const int x0   = wave * 16;
  const int kx0  = half * 2;

  v8f acc[4] = {v8f{}, v8f{}, v8f{}, v8f{}};
  const v2f* ap = (const v2f*)apack;

  for (int ci = 0; ci < 64; ++ci) {
#pragma unroll
    for (int ky = 0; ky < 3; ++ky) {
      const int row = ci * 3 + ky;
      const int col = row * 132 + x0 + n + kx0;
      v2f bf;
      bf.x = Xs[col];
      bf.y = Xs[col + 1];
      const v2f* a4 = ap + (size_t)row * 128 + lane;   // 4 co-tiles, 32 lanes each
      v2f a0 = a4[0];
      v2f a1 = a4[32];
      v2f a2 = a4[64];
      v2f a3 = a4[96];
      acc[0] = __builtin_amdgcn_wmma_f32_16x16x4_f32(false, a0, false, bf, (short)0, acc[0], false, false);
      acc[1] = __builtin_amdgcn_wmma_f32_16x16x4_f32(false, a1, false, bf, (short)0, acc[1], false, false);
      acc[2] = __builtin_amdgcn_wmma_f32_16x16x4_f32(false, a2, false, bf, (short)0, acc[2], false, false);
      acc[3] = __builtin_amdgcn_wmma_f32_16x16x4_f32(false, a3, false, bf, (short)0, acc[3], false, false);
    }
  }

  // Epilogue: bias + LeakyReLU, store NHWC (each lane writes 4x8 contiguous floats)
  const int xg = x0 + n;
  float* hb = h + (size_t)(((b * 128 + y) * 128 + xg)) * 64;
#pragma unroll
  for (int t = 0; t < 4; ++t) {
    const int cb = t * 16 + half * 8;
#pragma unroll
    for (int r = 0; r < 8; ++r) {
      float v = acc[t][r] + conv_b[cb + r];
      v = v > 0.0f ? v : NEG_SLOPE * v;
      hb[cb + r] = v;
    }
  }
}

// ---------------------------------------------------------------------------
// Kernel 2: gather kernel_seq rows from NHWC h and GEMM (768 -> 24) via WMMA.
// One wave per 16 (k,b) points; 192 k-steps; OUT padded 24 -> 32 with zero A.
// out layout: (K, B, 24) flat.
// ---------------------------------------------------------------------------
__global__ __launch_bounds__(256) void gather_gemm(const float* __restrict__ h,
                                                   const int* __restrict__ pos,
                                                   const float* __restrict__ apack2,
                                                   const float* __restrict__ pred_b,
                                                   float* __restrict__ out) {
  const int tid  = threadIdx.x;
  const int lane = tid & 31;
  const int wave = tid >> 5;
  const int tile = blockIdx.x * 8 + wave;            // 0..1999
  const int q    = lane & 15;
  const int half = lane >> 4;
  const int kk   = half * 2;
  const int nidx = tile * 16 + q;                    // 0..31999
  const int kp   = nidx >> 5;                        // point id (B = 32)
  const int b    = nidx & 31;

  v8f acc[2] = {v8f{}, v8f{}};
  const v2f* ap = (const v2f*)apack2;
  const int* pp = pos + (size_t)(kp * 32 + b) * 12 * 2;

  for (int l = 0; l < 12; ++l) {
    int px = pp[l * 2 + 0];
    int py = pp[l * 2 + 1];
    px = min(max(px, 0), 127);
    py = min(max(py, 0), 127);
    const float* base = h + ((size_t)(((b * 128 + py) * 128) + px) << 6);
#pragma unroll
    for (int s = 0; s < 16; ++s) {
      v2f bf = *(const v2f*)(base + s * 4 + kk);     // 8B-aligned in NHWC
      const int js = l * 16 + s;
      v2f a0 = ap[(js * 2 + 0) * 32 + lane];
      v2f a1 = ap[(js * 2 + 1) * 32 + lane];
      acc[0] = __builtin_amdgcn_wmma_f32_16x16x4_f32(false, a0, false, bf, (short)0, acc[0], false, false);
      acc[1] = __builtin_amdgcn_wmma_f32_16x16x4_f32(false, a1, false, bf, (short)0, acc[1], false, false);
    }
  }

#pragma unroll
  for (int t = 0; t < 2; ++t) {
#pragma unroll
    for (int r = 0; r < 8; ++r) {
      const int o = t * 16 + half * 8 + r;
      if (o < 24) out[(size_t)nidx * 24 + o] = acc[t][r] + pred_b[o];
    }
  }
}

// ---------------------------------------------------------------------------
extern "C" void kernel_launch(void* const* d_in, const int* in_sizes, int n_in,
                              void* d_out, int out_size, void* d_ws, size_t ws_size,
                              hipStream_t stream) {
  const float* x      = (const float*)d_in[0];
  const int*   pos    = (const int*)d_in[1];
  const float* conv_w = (const float*)d_in[2];
  const float* conv_b = (const float*)d_in[3];
  const float* pred_w = (const float*)d_in[4];
  const float* pred_b = (const float*)d_in[5];
  float* out = (float*)d_out;

  char* ws = (char*)d_ws;
  const size_t H_BYTES  = (size_t)32 * 128 * 128 * 64 * 4;   // 134,217,728 (h NHWC)
  const size_t AP1_BYTES = (size_t)64 * 3 * 4 * 32 * 8;      // 196,608
  float* h_buf   = (float*)ws;
  float* apack1  = (float*)(ws + H_BYTES);
  float* apack2  = (float*)(ws + H_BYTES + AP1_BYTES);

  pack_conv_w<<<96, 256, 0, stream>>>(conv_w, apack1);
  pack_pred_w<<<48, 256, 0, stream>>>(pred_w, apack2);
  conv_lrelu<<<32 * 128, 256, 0, stream>>>(x, conv_b, apack1, h_buf);
  gather_gemm<<<2000 / 8, 256, 0, stream>>>(h_buf, pos, apack2, pred_b, out);
}